// CustomMLPLayer_74526272520565
// MI455X (gfx1250) — compile-verified
//
#include <hip/hip_runtime.h>
#include <stdint.h>

// Problem constants (from reference)
#define S_TOK  2048          // tokens
#define H_DIM  11008         // hidden size
#define D_DIM  4096          // output dim
#define K_TOKN 2201          // int(11008 * 0.2) per-token top-k
#define N_CORE 4403          // int(11008 * 0.4) core neurons

// Workspace layout (bytes)
#define WS_THR 0             // float[2048]
#define WS_CNT 8192          // uint[11008]
#define WS_IDX 52224         // int[N_CORE]

// ---------------------------------------------------------------------------
// CDNA5 async global->LDS copy support (ASYNCcnt path), with sync fallback.
// ---------------------------------------------------------------------------
#if defined(__has_builtin)
#if __has_builtin(__builtin_amdgcn_global_load_async_to_lds_b32) && \
    __has_builtin(__builtin_amdgcn_s_wait_asynccnt)
#define USE_ASYNC_LDS 1
#endif
#endif
#ifndef USE_ASYNC_LDS
#define USE_ASYNC_LDS 0
#endif

typedef __attribute__((address_space(1))) int gint_t;   // global (prints as __device__ int*)
typedef __attribute__((address_space(3))) int lint_t;   // LDS    (prints as __shared__ int*)

__device__ __forceinline__ void lds_gather_b32(float* lds_dst, const float* gsrc) {
#if USE_ASYNC_LDS
  // AS1 pointer is bit-identical to the flat address; AS3 pointer is the low
  // 32 bits of the flat LDS-aperture address.
  __builtin_amdgcn_global_load_async_to_lds_b32(
      (gint_t*)(uintptr_t)gsrc,
      (lint_t*)(uint32_t)(uintptr_t)lds_dst,
      0, 0);
#else
  *lds_dst = *gsrc;
#endif
}

__device__ __forceinline__ void async_wait_all() {
#if USE_ASYNC_LDS
  __builtin_amdgcn_s_wait_asynccnt(0);
#endif
}

// ---------------------------------------------------------------------------
// Order-preserving float <-> uint mapping (larger float <-> larger uint)
// ---------------------------------------------------------------------------
__device__ __forceinline__ uint32_t f2ord(float f) {
  uint32_t u = __float_as_uint(f);
  return (u & 0x80000000u) ? ~u : (u | 0x80000000u);
}
__device__ __forceinline__ float ord2f(uint32_t u) {
  uint32_t v = (u & 0x80000000u) ? (u & 0x7fffffffu) : ~u;
  return __uint_as_float(v);
}

// ---------------------------------------------------------------------------
// Kernel 1: exact per-token k-th largest via 4-pass radix select.
// One 256-thread workgroup per token.
// ---------------------------------------------------------------------------
__global__ __launch_bounds__(256) void topk_threshold_kernel(
    const float* __restrict__ x, float* __restrict__ thr) {
  __shared__ uint32_t hist[256];
  __shared__ uint32_t sPrefix;
  __shared__ int sK;

  const int token = blockIdx.x;
  const float* row = x + (size_t)token * H_DIM;

  if (threadIdx.x == 0) { sPrefix = 0u; sK = K_TOKN; }
  __syncthreads();

  for (int pass = 0; pass < 4; ++pass) {
    const int shift = 24 - 8 * pass;
    hist[threadIdx.x] = 0u;
    __syncthreads();
    const uint32_t prefix = sPrefix;
    for (int i = threadIdx.x; i < H_DIM; i += 256) {
      uint32_t u = f2ord(row[i]);
      bool match = (pass == 0) || (((u ^ prefix) >> (shift + 8)) == 0u);
      if (match) atomicAdd(&hist[(u >> shift) & 255u], 1u);
    }
    __syncthreads();
    if (threadIdx.x == 0) {
      int k = sK;
      int cum = 0;
      int d = 255;
      for (; d > 0; --d) {
        int nc = cum + (int)hist[d];
        if (nc >= k) break;
        cum = nc;
      }
      sK = k - cum;                       // remaining rank inside digit d
      sPrefix = prefix | ((uint32_t)d << shift);
    }
    __syncthreads();
  }
  if (threadIdx.x == 0) thr[token] = ord2f(sPrefix);
}

// ---------------------------------------------------------------------------
// Kernel 2a: zero the vote counters.
// ---------------------------------------------------------------------------
__global__ __launch_bounds__(256) void zero_counts_kernel(uint32_t* __restrict__ cnt) {
  int i = blockIdx.x * 256 + threadIdx.x;
  if (i < H_DIM) cnt[i] = 0u;
}

// ---------------------------------------------------------------------------
// Kernel 2b: vote counts. Each thread owns one column over a 256-token slab,
// accumulates locally, one global atomic per thread at the end.
// grid = (ceil(H/256), S/256)
// ---------------------------------------------------------------------------
__global__ __launch_bounds__(256) void count_kernel(
    const float* __restrict__ x, const float* __restrict__ thr,
    uint32_t* __restrict__ cnt) {
  __shared__ float sThr[256];
  const int col = blockIdx.x * 256 + threadIdx.x;
  const int s0  = blockIdx.y * 256;
  sThr[threadIdx.x] = thr[s0 + threadIdx.x];
  __syncthreads();
  if (col >= H_DIM) return;
  const float* p = x + (size_t)s0 * H_DIM + col;
  uint32_t acc = 0u;
  for (int s = 0; s < 256; ++s)
    acc += (p[(size_t)s * H_DIM] >= sThr[s]) ? 1u : 0u;
  atomicAdd(&cnt[col], acc);
}

// ---------------------------------------------------------------------------
// Kernel 3: pick the N_CORE most-voted neurons, ties broken by lowest index
// (matches jax.lax.top_k), and emit indices in ascending (sorted) order.
// Single 1024-thread workgroup.
// ---------------------------------------------------------------------------
__global__ __launch_bounds__(1024) void select_core_kernel(
    const uint32_t* __restrict__ cnt, int* __restrict__ core_idx) {
  __shared__ uint32_t hist[S_TOK + 1];   // counts can be 0..2048
  __shared__ uint32_t scanA[1024];
  __shared__ uint32_t scanE[1024];
  __shared__ int sCstar, sQ;
  __shared__ uint32_t baseA, baseE;

  const int tid = threadIdx.x;
  for (int i = tid; i <= S_TOK; i += 1024) hist[i] = 0u;
  __syncthreads();
  for (int i = tid; i < H_DIM; i += 1024) {
    uint32_t c = cnt[i];
    if (c > (uint32_t)S_TOK) c = S_TOK;
    atomicAdd(&hist[c], 1u);
  }
  __syncthreads();
  if (tid == 0) {
    // find count threshold c*: #{cnt > c*} < N_CORE <= #{cnt >= c*}
    int cum = 0, cstar = 0, above = 0;
    for (int c = S_TOK; c >= 0; --c) {
      int nc = cum + (int)hist[c];
      if (nc >= N_CORE) { cstar = c; above = cum; break; }
      cum = nc;
    }
    sCstar = cstar;
    sQ = N_CORE - above;   // number of ties (count == c*) to keep, lowest-index first
    baseA = 0u; baseE = 0u;
  }
  __syncthreads();
  const int cstar = sCstar;
  const uint32_t q = (uint32_t)sQ;

  const int nChunks = (H_DIM + 1023) / 1024;
  for (int ch = 0; ch < nChunks; ++ch) {
    const int n = ch * 1024 + tid;
    uint32_t c = (n < H_DIM) ? cnt[n] : 0u;
    uint32_t isA = (n < H_DIM && (int)c > cstar) ? 1u : 0u;
    uint32_t isE = (n < H_DIM && (int)c == cstar) ? 1u : 0u;
    scanA[tid] = isA; scanE[tid] = isE;
    __syncthreads();
    for (int off = 1; off < 1024; off <<= 1) {   // Hillis-Steele inclusive scan
      uint32_t a = (tid >= off) ? scanA[tid - off] : 0u;
      uint32_t e = (tid >= off) ? scanE[tid - off] : 0u;
      __syncthreads();
      scanA[tid] += a; scanE[tid] += e;
      __syncthreads();
    }
    const uint32_t prefA = baseA + scanA[tid] - isA;   // exclusive, global
    const uint32_t prefE = baseE + scanE[tid] - isE;
    if (isA) {
      uint32_t eqBefore = (prefE < q) ? prefE : q;
      core_idx[prefA + eqBefore] = n;
    } else if (isE && prefE < q) {
      core_idx[prefA + prefE] = n;
    }
    __syncthreads();
    if (tid == 0) { baseA += scanA[1023]; baseE += scanE[1023]; }
    __syncthreads();
  }
}

// ---------------------------------------------------------------------------
// Kernel 4: fused-gather WMMA GEMM (f32 x f32 -> f32, full precision).
//   out[s, d] = sum_k x[s, idx[k]] * W[d, idx[k]]
// WG tile 128x128, BK=32, double-buffered LDS staged with async global->LDS
// gathers. [m][k] LDS layout (pitch 34, conflict-free) so each WMMA fragment
// is one aligned ds_load_b64.
// 8 waves, each wave: 64x32 out = 4x2 subtiles of v_wmma_f32_16x16x4_f32.
// ---------------------------------------------------------------------------
typedef float v2f __attribute__((ext_vector_type(2)));
typedef float v8f __attribute__((ext_vector_type(8)));

#define BM  128
#define BN  128
#define BK  32
#define BKP (BK + 2)   // LDS row pitch (words); 8B-aligned, bank-conflict-free

__shared__ float As[2][BM][BKP];
__shared__ float Bs[2][BM][BKP];

__global__ __launch_bounds__(256) void core_gemm_kernel(
    const float* __restrict__ x, const float* __restrict__ W,
    const int* __restrict__ core_idx, float* __restrict__ out) {
  const int tid   = threadIdx.x;
  const int lane  = tid & 31;
  const int wave  = tid >> 5;       // 0..7
  const int wm    = wave >> 2;      // 0..1 -> 64-row slab
  const int wn    = wave & 3;       // 0..3 -> 32-col slab
  const int half  = lane >> 4;      // 0/1 (K pair select per ISA layout)
  const int l16   = lane & 15;
  const int sBase = blockIdx.y * BM;
  const int dBase = blockIdx.x * BN;

  // Staging geometry: thread always stages column k = tid&31 for rows
  // m = (tid>>5) + 8*i, i = 0..15 — one gathered neuron index per thread/tile.
  const int kst = tid & 31;
  const int mst = tid >> 5;

  v8f acc[4][2];
  const v8f vzero = {0.f, 0.f, 0.f, 0.f, 0.f, 0.f, 0.f, 0.f};
#pragma unroll
  for (int i = 0; i < 4; ++i)
#pragma unroll
    for (int j = 0; j < 2; ++j) acc[i][j] = vzero;

  const int nTiles = (N_CORE + BK - 1) / BK;

  // ---- stage one K-tile into buffer `buf` (async when available) ----
  auto stage = [&](int t, int buf) {
    const int kk  = t * BK + kst;
    const int col = (kk < N_CORE) ? core_idx[kk] : -1;
    if (col >= 0) {
      const float* xp = x + (size_t)(sBase + mst) * H_DIM + col;
      const float* wp = W + (size_t)(dBase + mst) * H_DIM + col;
#pragma unroll
      for (int i = 0; i < 16; ++i) {
        const int m = mst + 8 * i;
        lds_gather_b32(&As[buf][m][kst], xp + (size_t)(8 * i) * H_DIM);
        lds_gather_b32(&Bs[buf][m][kst], wp + (size_t)(8 * i) * H_DIM);
      }
    } else {
#pragma unroll
      for (int i = 0; i < 16; ++i) {
        const int m = mst + 8 * i;
        As[buf][m][kst] = 0.f;
        Bs[buf][m][kst] = 0.f;
      }
    }
  };

  stage(0, 0);

  for (int t = 0; t < nTiles; ++t) {
    const int cur = t & 1;
    async_wait_all();     // my stage(t) writes to LDS are complete
    __syncthreads();      // everyone's stage(t) complete; buf cur^1 free
    if (t + 1 < nTiles) stage(t + 1, cur ^ 1);   // overlaps with compute below

#pragma unroll
    for (int kk = 0; kk < BK; kk += 4) {
      const int ka = kk + 2 * half;   // this lane-half's K pair
      v2f a[4], b[2];
#pragma unroll
      for (int i = 0; i < 4; ++i) {
        const int m = wm * 64 + i * 16 + l16;
        a[i] = *(const v2f*)&As[cur][m][ka];     // ds_load_b64
      }
#pragma unroll
      for (int j = 0; j < 2; ++j) {
        const int n = wn * 32 + j * 16 + l16;
        b[j] = *(const v2f*)&Bs[cur][n][ka];     // ds_load_b64
      }
#pragma unroll
      for (int i = 0; i < 4; ++i)
#pragma unroll
        for (int j = 0; j < 2; ++j)
          acc[i][j] = __builtin_amdgcn_wmma_f32_16x16x4_f32(
              false, a[i], false, b[j], (short)0, acc[i][j], false, false);
    }
  }

  // Write back: C/D layout — VGPR r: lanes 0-15 M=r, lanes 16-31 M=8+r.
#pragma unroll
  for (int i = 0; i < 4; ++i) {
    const int mBase = sBase + wm * 64 + i * 16 + 8 * half;
#pragma unroll
    for (int j = 0; j < 2; ++j) {
      const int n = dBase + wn * 32 + j * 16 + l16;
#pragma unroll
      for (int r = 0; r < 8; ++r)
        out[(size_t)(mBase + r) * D_DIM + n] = acc[i][j][r];
    }
  }
}

// ---------------------------------------------------------------------------
// Launch
// ---------------------------------------------------------------------------
extern "C" void kernel_launch(void* const* d_in, const int* in_sizes, int n_in,
                              void* d_out, int out_size, void* d_ws, size_t ws_size,
                              hipStream_t stream) {
  const float* x = (const float*)d_in[0];   // [1, 2048, 11008] f32
  const float* W = (const float*)d_in[1];   // [4096, 11008] f32
  float* out = (float*)d_out;               // [1, 2048, 4096] f32

  char* ws = (char*)d_ws;
  float*    thr      = (float*)(ws + WS_THR);
  uint32_t* cnt      = (uint32_t*)(ws + WS_CNT);
  int*      core_idx = (int*)(ws + WS_IDX);

  topk_threshold_kernel<<<S_TOK, 256, 0, stream>>>(x, thr);
  zero_counts_kernel<<<(H_DIM + 255) / 256, 256, 0, stream>>>(cnt);
  count_kernel<<<dim3((H_DIM + 255) / 256, S_TOK / 256), 256, 0, stream>>>(x, thr, cnt);
  select_core_kernel<<<1, 1024, 0, stream>>>(cnt, core_idx);
  core_gemm_kernel<<<dim3(D_DIM / BN, S_TOK / BM), 256, 0, stream>>>(x, W, core_idx, out);
}